// SAGEConvWithCV_5282809774188
// MI455X (gfx1250) — compile-verified
//
#include <hip/hip_runtime.h>

#define N_SRC   100000
#define N_DST   50000
#define N_EDGES 800000
#define D       128
#define D_OUT   128
#define K_TOT   (2 * D)   // 256

typedef __attribute__((ext_vector_type(2))) float v2f;
typedef __attribute__((ext_vector_type(8))) float v8f;

// ---------------------------------------------------------------------------
// Kernel 1: zero the scratch accumulators (summed[N_DST*D], deg[N_DST])
// ---------------------------------------------------------------------------
__global__ void zero_ws_kernel(float* __restrict__ summed, float* __restrict__ deg) {
    int i = blockIdx.x * blockDim.x + threadIdx.x;
    if (i < N_DST * D) summed[i] = 0.0f;
    if (i < N_DST)     deg[i]    = 0.0f;
}

// ---------------------------------------------------------------------------
// Kernel 2: per-edge gather of (H_src - HBar_src) and scatter-add to dst.
// 32 threads per edge, float4 per thread -> 4 atomic f32 adds (L2-resident).
// ---------------------------------------------------------------------------
__global__ void edge_scatter_kernel(const float* __restrict__ H_src,
                                    const float* __restrict__ HBar_src,
                                    const int*   __restrict__ src_idx,
                                    const int*   __restrict__ dst_idx,
                                    float* __restrict__ summed,
                                    float* __restrict__ deg) {
    unsigned tid = blockIdx.x * blockDim.x + threadIdx.x;
    unsigned e = tid >> 5;          // edge index (32 lanes per edge)
    unsigned c = tid & 31;          // feature chunk (4 floats each)
    if (e >= N_EDGES) return;

    int s = src_idx[e];
    int d = dst_idx[e];
    int f = c * 4;

    const float4 a = *reinterpret_cast<const float4*>(H_src    + (size_t)s * D + f);
    const float4 b = *reinterpret_cast<const float4*>(HBar_src + (size_t)s * D + f);

    float* dstp = summed + (size_t)d * D + f;
    atomicAdd(dstp + 0, a.x - b.x);
    atomicAdd(dstp + 1, a.y - b.y);
    atomicAdd(dstp + 2, a.z - b.z);
    atomicAdd(dstp + 3, a.w - b.w);

    if (c == 0) atomicAdd(&deg[d], 1.0f);
}

// ---------------------------------------------------------------------------
// Kernel 3: finalize h_neigh in place: summed <- agg + summed / max(deg,1)
// ---------------------------------------------------------------------------
__global__ void finalize_kernel(const float* __restrict__ agg,
                                float* __restrict__ summed,
                                const float* __restrict__ deg) {
    int i = blockIdx.x * blockDim.x + threadIdx.x;
    if (i >= N_DST * D) return;
    int row = i >> 7;               // / D
    float inv = 1.0f / fmaxf(deg[row], 1.0f);
    summed[i] = agg[i] + summed[i] * inv;
}

// ---------------------------------------------------------------------------
// Kernel 4: out = relu([H_dst | hneigh] @ W + b) via V_WMMA_F32_16X16X4_F32.
// Block = 256 threads = 8 waves. Block covers 16 dst rows x all 128 out cols;
// wave w owns the 16-wide column tile w. X tile staged in LDS (stride 260 to
// avoid bank conflicts). 64 K-steps of 4.
// ---------------------------------------------------------------------------
#define XS_STRIDE 260

__global__ void gemm_wmma_kernel(const float* __restrict__ H_dst,
                                 const float* __restrict__ hneigh,
                                 const float* __restrict__ W,      // [256,128]
                                 const float* __restrict__ bias,   // [128]
                                 float* __restrict__ out) {        // [N_DST,128]
    __shared__ float Xs[16 * XS_STRIDE];

    const int rowtile = blockIdx.x;           // 16 rows per block (3125 blocks)
    // --- cooperative load of X tile: 16 rows x 256 cols ---------------------
    {
        int m  = threadIdx.x >> 4;            // row within tile (16 loaders/row)
        int c0 = (threadIdx.x & 15) * 16;     // 16 consecutive floats per thread
        int grow = rowtile * 16 + m;
        const float* srcrow = (c0 < D) ? (H_dst  + (size_t)grow * D + c0)
                                       : (hneigh + (size_t)grow * D + (c0 - D));
        #pragma unroll
        for (int j = 0; j < 16; j += 4) {
            float4 v = *reinterpret_cast<const float4*>(srcrow + j);
            *reinterpret_cast<float4*>(&Xs[m * XS_STRIDE + c0 + j]) = v;
        }
    }
    __syncthreads();

    const int lane  = threadIdx.x & 31;
    const int wavei = threadIdx.x >> 5;       // column tile 0..7
    const int nn    = lane & 15;              // A: row M   /  B: col N
    const int half  = lane >> 4;              // selects K pair within step

    v8f acc = {0.f, 0.f, 0.f, 0.f, 0.f, 0.f, 0.f, 0.f};

    const int colbase = wavei * 16 + nn;
    #pragma unroll 8
    for (int kk = 0; kk < K_TOT; kk += 4) {
        int k0 = kk + half * 2;
        v2f a, b;
        a.x = Xs[nn * XS_STRIDE + k0];
        a.y = Xs[nn * XS_STRIDE + k0 + 1];
        b.x = W[(size_t)k0 * D_OUT + colbase];
        b.y = W[(size_t)(k0 + 1) * D_OUT + colbase];
        acc = __builtin_amdgcn_wmma_f32_16x16x4_f32(
            /*neg_a=*/false, a, /*neg_b=*/false, b,
            /*c_mod=*/(short)0, acc, /*reuse_a=*/false, /*reuse_b=*/false);
    }

    // --- epilogue: bias + relu + store (C/D layout per ISA 7.12.2) ----------
    float bv = bias[colbase];
    #pragma unroll
    for (int r = 0; r < 8; ++r) {
        int row = rowtile * 16 + r + half * 8;
        float v = fmaxf(acc[r] + bv, 0.0f);
        out[(size_t)row * D_OUT + colbase] = v;
    }
}

// ---------------------------------------------------------------------------
extern "C" void kernel_launch(void* const* d_in, const int* in_sizes, int n_in,
                              void* d_out, int out_size, void* d_ws, size_t ws_size,
                              hipStream_t stream) {
    const float* H_src    = (const float*)d_in[0];
    const float* H_dst    = (const float*)d_in[1];
    const float* HBar_src = (const float*)d_in[2];
    const float* agg      = (const float*)d_in[3];
    const float* W        = (const float*)d_in[4];
    const float* bias     = (const float*)d_in[5];
    const int*   src_idx  = (const int*)d_in[6];
    const int*   dst_idx  = (const int*)d_in[7];
    float*       out      = (float*)d_out;

    float* summed = (float*)d_ws;                 // N_DST * D floats (25.6 MB)
    float* deg    = summed + (size_t)N_DST * D;   // N_DST floats

    (void)in_sizes; (void)n_in; (void)out_size; (void)ws_size;

    // 1) zero accumulators
    {
        int total = N_DST * D;
        zero_ws_kernel<<<(total + 255) / 256, 256, 0, stream>>>(summed, deg);
    }
    // 2) edge scatter: 32 threads/edge
    {
        long long threads = (long long)N_EDGES * 32;
        int blocks = (int)((threads + 255) / 256);   // 100000
        edge_scatter_kernel<<<blocks, 256, 0, stream>>>(H_src, HBar_src,
                                                        src_idx, dst_idx,
                                                        summed, deg);
    }
    // 3) finalize h_neigh in place
    {
        int total = N_DST * D;
        finalize_kernel<<<(total + 255) / 256, 256, 0, stream>>>(agg, summed, deg);
    }
    // 4) WMMA GEMM + bias + relu
    {
        gemm_wmma_kernel<<<N_DST / 16, 256, 0, stream>>>(H_dst, summed, W, bias, out);
    }
}